// Model_7705171329776
// MI455X (gfx1250) — compile-verified
//
#include <hip/hip_runtime.h>
#include <hip/hip_bf16.h>
#include <math.h>

// Problem constants (match reference)
constexpr int kB = 16;     // batch
constexpr int kN = 4096;   // nodes
constexpr int kC = 128;    // channels
constexpr int kE = 12288;  // incidences
constexpr int kM = 64;     // hyperedges
constexpr int kBC = kB * kC;       // 2048
constexpr int kNB = kN * kB;       // 65536
constexpr int kMB = kM * kB;       // 1024
constexpr float kNegSlope = 0.2f;
constexpr float kMargin = 4.2f;

typedef float v2f __attribute__((ext_vector_type(2)));
typedef float v8f __attribute__((ext_vector_type(8)));

// ---------- monotone float<->u32 key for order-independent atomic max ----------
__device__ inline unsigned fkey(float f) {
    int i = __float_as_int(f);
    return (unsigned)(i >= 0 ? (i | 0x80000000) : ~i);
}
__device__ inline float funkey(unsigned k) {
    int i = (k & 0x80000000u) ? (int)(k & 0x7FFFFFFFu) : ~(int)k;
    return __int_as_float(i);
}

// ---------- K0: zero/init accumulated buffers ----------
__global__ __launch_bounds__(256) void init_kernel(float* __restrict__ out,
                                                   unsigned* __restrict__ segmax,
                                                   float* __restrict__ segsum,
                                                   int* __restrict__ deg) {
    int i = blockIdx.x * 256 + threadIdx.x;
    if (i < kB * kN * kC) out[i] = 0.0f;
    if (i < kNB) { segmax[i] = 0u; segsum[i] = 0.0f; }
    if (i < kN) deg[i] = 0;
}

// ---------- K1: fp32 WMMA GEMM  x[B*N,C] @ W[C,C] -> x1[N,B,C] ----------
// One wave per 16-row stripe, 8 column tiles of 16 -> 256 v_wmma_f32_16x16x4_f32 per wave.
__global__ __launch_bounds__(256) void gemm_xw_wmma(const float* __restrict__ x,
                                                    const float* __restrict__ w,
                                                    float* __restrict__ x1) {
    // W staged in LDS, pair-of-k layout: wlds[(k/2)*256 + (n ^ sw(k))*2 + (k&1)]
    // sw(k) = ((k>>1)&1)<<4 kills bank conflicts between lane halves.
    __shared__ float wlds[kC * kC];  // 64 KB
    const int tid = threadIdx.x;
    for (int i = tid * 4; i < kC * kC; i += 256 * 4) {
        float4 v = *(const float4*)(w + i);
        int k = i >> 7, n = i & 127;
        int sw = ((k >> 1) & 1) << 4;
        int base = ((k >> 1) << 8) + (k & 1);
        wlds[base + ((n ^ sw) << 1)]       = v.x;
        wlds[base + (((n + 1) ^ sw) << 1)] = v.y;
        wlds[base + (((n + 2) ^ sw) << 1)] = v.z;
        wlds[base + (((n + 3) ^ sw) << 1)] = v.w;
    }
    __syncthreads();

    const int wave = tid >> 5, lane = tid & 31;
    const int rt = blockIdx.x * 8 + wave;   // row tile id (0..4095)
    const int r0 = rt << 4;
    const int row = r0 + (lane & 15);
    const int kh  = (lane >> 4) << 1;       // 0 or 2 (A K-offset per lane half)
    const int csw = (lane >> 4) << 4;       // B column swizzle per lane half
    const int c0  = lane & 15;

    const float* xrow = x + (size_t)row * kC;
    v8f acc[8];
    v8f zero = {0.f, 0.f, 0.f, 0.f, 0.f, 0.f, 0.f, 0.f};
#pragma unroll
    for (int i = 0; i < 8; i++) acc[i] = zero;

#pragma unroll 2
    for (int kb = 0; kb < kC; kb += 4) {
        float2 av = *(const float2*)(xrow + kb + kh);
        v2f a; a.x = av.x; a.y = av.y;
        const float2* wp = (const float2*)&wlds[((kb + kh) >> 1) << 8];
#pragma unroll
        for (int ct = 0; ct < 8; ct++) {
            float2 bv = wp[((ct << 4) + c0) ^ csw];
            v2f b; b.x = bv.x; b.y = bv.y;
            acc[ct] = __builtin_amdgcn_wmma_f32_16x16x4_f32(
                false, a, false, b, (short)0, acc[ct], false, false);
        }
    }

    // C/D layout: VGPR g, lanes 0-15 -> (M=g, N=lane); lanes 16-31 -> (M=g+8, N=lane-16)
    const int batch  = r0 >> 12;            // constant within a 16-row tile
    const int rowadd = (lane >> 4) << 3;
#pragma unroll
    for (int ct = 0; ct < 8; ct++) {
#pragma unroll
        for (int g = 0; g < 8; g++) {
            int n = (r0 + g + rowadd) & (kN - 1);
            x1[(size_t)((n << 4) + batch) * kC + (ct << 4) + c0] = acc[ct][g];
        }
    }
}

// ---------- K2: edge_sums[m] = sum over incidences of x1[nodes[e]]  (deterministic) ----------
__global__ __launch_bounds__(256) void edge_sums_kernel(const float* __restrict__ x1,
                                                        const int* __restrict__ nodes,
                                                        const int* __restrict__ edges,
                                                        float* __restrict__ es) {
    const int m = blockIdx.x, tid = threadIdx.x;
    __shared__ int se[256], sn[256];
    float acc[8] = {0, 0, 0, 0, 0, 0, 0, 0};
    for (int base = 0; base < kE; base += 256) {
        se[tid] = edges[base + tid];
        sn[tid] = nodes[base + tid];
        __syncthreads();
        for (int i = 0; i < 256; i++) {
            if (se[i] == m) {
                const float* src = x1 + (size_t)sn[i] * kBC;
#pragma unroll
                for (int j = 0; j < 8; j++) acc[j] += src[tid + j * 256];
            }
        }
        __syncthreads();
    }
#pragma unroll
    for (int j = 0; j < 8; j++) es[(size_t)m * kBC + tid + j * 256] = acc[j];
}

// ---------- K3: y1[n,b]=x1.att1 ; y2[m,b]=es.att2 ; sq[m,b]=|es|^2  (one wave per dot) ----------
__global__ __launch_bounds__(256) void dots_kernel(const float* __restrict__ x1,
                                                   const float* __restrict__ es,
                                                   const float* __restrict__ att,
                                                   float* __restrict__ y1,
                                                   float* __restrict__ y2,
                                                   float* __restrict__ sq) {
    const int lane = threadIdx.x & 31;
    const int job = blockIdx.x * 8 + (threadIdx.x >> 5);
    if (job >= kNB + 2 * kMB) return;
    const float* src;
    const float* vec;
    if (job < kNB)             { src = x1 + (size_t)job * kC;               vec = att; }
    else if (job < kNB + kMB)  { src = es + (size_t)(job - kNB) * kC;       vec = att + kC; }
    else                       { src = es + (size_t)(job - kNB - kMB) * kC; vec = src; }
    float4 s = *(const float4*)(src + lane * 4);
    float4 v = *(const float4*)(vec + lane * 4);
    float p = s.x * v.x + s.y * v.y + s.z * v.z + s.w * v.w;
#pragma unroll
    for (int off = 16; off >= 1; off >>= 1) p += __shfl_xor(p, off, 32);
    if (lane == 0) {
        if (job < kNB)            y1[job] = p;
        else if (job < kNB + kMB) y2[job - kNB] = p;
        else                      sq[job - kNB - kMB] = p;
    }
}

// ---------- K4a: logits + leaky relu + segment max (+ node degree) ----------
__global__ __launch_bounds__(256) void alpha_logits_kernel(const int* __restrict__ nodes,
                                                           const int* __restrict__ edges,
                                                           const float* __restrict__ y1,
                                                           const float* __restrict__ y2,
                                                           float* __restrict__ alpha,
                                                           unsigned* __restrict__ segmax,
                                                           int* __restrict__ deg) {
    int idx = blockIdx.x * 256 + threadIdx.x;   // over E*B
    int e = idx >> 4, b = idx & 15;
    int n = nodes[e];
    float l = y1[n * kB + b] + y2[edges[e] * kB + b];
    l = l > 0.0f ? l : kNegSlope * l;
    alpha[idx] = l;
    atomicMax(&segmax[n * kB + b], fkey(l));
    if (b == 0) atomicAdd(&deg[n], 1);
}

// ---------- K4b: exp(l - max) + segment sum ----------
__global__ __launch_bounds__(256) void alpha_exp_kernel(const int* __restrict__ nodes,
                                                        float* __restrict__ alpha,
                                                        const unsigned* __restrict__ segmax,
                                                        float* __restrict__ segsum) {
    int idx = blockIdx.x * 256 + threadIdx.x;
    int e = idx >> 4, b = idx & 15;
    int s = nodes[e] * kB + b;
    float ex = __expf(alpha[idx] - funkey(segmax[s]));
    alpha[idx] = ex;
    atomicAdd(&segsum[s], ex);
}

// ---------- K4c: normalize ----------
__global__ __launch_bounds__(256) void alpha_norm_kernel(const int* __restrict__ nodes,
                                                         float* __restrict__ alpha,
                                                         const float* __restrict__ segsum) {
    int idx = blockIdx.x * 256 + threadIdx.x;
    int e = idx >> 4, b = idx & 15;
    alpha[idx] = alpha[idx] / (segsum[nodes[e] * kB + b] + 1e-16f);
}

// ---------- K5: out_e[m] = Bn[m] * sum_e alpha[e,b]*x1[nodes[e]]  (deterministic) ----------
__global__ __launch_bounds__(256) void out_e_kernel(const float* __restrict__ x1,
                                                    const int* __restrict__ nodes,
                                                    const int* __restrict__ edges,
                                                    const float* __restrict__ alpha,
                                                    float* __restrict__ oe) {
    const int m = blockIdx.x, tid = threadIdx.x;
    __shared__ int se[256], sn[256];
    float acc[8] = {0, 0, 0, 0, 0, 0, 0, 0};
    int cnt = 0;
    for (int base = 0; base < kE; base += 256) {
        se[tid] = edges[base + tid];
        sn[tid] = nodes[base + tid];
        __syncthreads();
        for (int i = 0; i < 256; i++) {
            if (se[i] == m) {
                cnt++;
                const float* src = x1 + (size_t)sn[i] * kBC;
                const float* al = alpha + (size_t)(base + i) * kB;
#pragma unroll
                for (int j = 0; j < 8; j++) {
                    int idx = tid + j * 256;
                    acc[j] += al[idx >> 7] * src[idx];
                }
            }
        }
        __syncthreads();
    }
    float bn = cnt > 0 ? 1.0f / (float)cnt : 0.0f;
#pragma unroll
    for (int j = 0; j < 8; j++) oe[(size_t)m * kBC + tid + j * 256] = acc[j] * bn;
}

// ---------- K6: out[b,n,c] += D[n]*out_e[edges[e]]*alpha  (incidence-parallel scatter) ----------
__global__ __launch_bounds__(256) void scatter_out_kernel(const int* __restrict__ nodes,
                                                          const int* __restrict__ edges,
                                                          const float* __restrict__ alpha,
                                                          const float* __restrict__ oe,
                                                          const int* __restrict__ deg,
                                                          float* __restrict__ out) {
    int idx = blockIdx.x * 256 + threadIdx.x;   // over E*B*C
    int e = idx >> 11, r = idx & 2047;
    int b = r >> 7, c = r & 127;
    int n = nodes[e];
    float val = (float)deg[n] * oe[(size_t)edges[e] * kBC + r] * alpha[e * kB + b];
    atomicAdd(&out[(size_t)b * kN * kC + (size_t)n * kC + c], val);
}

// ---------- K7: pairwise hyperedge loss, lh[m*64+n] = mean_b(loss_item) ----------
__global__ __launch_bounds__(128) void pair_loss_kernel(const float* __restrict__ es,
                                                        const float* __restrict__ sq,
                                                        float* __restrict__ lh) {
    const int mm = blockIdx.x >> 6, nn = blockIdx.x & 63;
    const int t = threadIdx.x, b = t >> 3, ci = t & 7;
    const float* pm = es + (size_t)(mm * kB + b) * kC + ci * 16;
    const float* pn = es + (size_t)(nn * kB + b) * kC + ci * 16;
    float p = 0.0f;
#pragma unroll
    for (int c = 0; c < 16; c += 4) {
        float4 a = *(const float4*)(pm + c);
        float4 q = *(const float4*)(pn + c);
        p += a.x * q.x + a.y * q.y + a.z * q.z + a.w * q.w;
    }
    __shared__ float red[128];
    __shared__ float li[16];
    red[t] = p;
    __syncthreads();
    if (t < 16) {
        float inner = 0.0f;
        for (int k = 0; k < 8; k++) inner += red[t * 8 + k];
        float sm = sq[mm * kB + t], sn = sq[nn * kB + t];
        float cosv = inner / (sqrtf(sm) * sqrtf(sn));
        float d2 = sm + sn - 2.0f * inner;
        float dist = sqrtf(d2 > 0.0f ? d2 : 0.0f);
        float mg = kMargin - dist;
        mg = mg > 0.0f ? mg : 0.0f;
        li[t] = cosv * dist + (1.0f - cosv) * mg;
    }
    __syncthreads();
    if (t == 0) {
        float s = 0.0f;
        for (int k = 0; k < 16; k++) s += li[k];
        lh[blockIdx.x] = s * (1.0f / (float)kB);
    }
}

// ---------- K8: per-incidence sum of (x_i - x_j) for the constraint mean ----------
__global__ __launch_bounds__(256) void diff_kernel(const float* __restrict__ x1,
                                                   const float* __restrict__ es,
                                                   const int* __restrict__ nodes,
                                                   const int* __restrict__ edges,
                                                   float* __restrict__ pb) {
    const int e = blockIdx.x, t = threadIdx.x;
    const float* xi = x1 + (size_t)nodes[e] * kBC;
    const float* xj = es + (size_t)edges[e] * kBC;
    float p = 0.0f;
#pragma unroll
    for (int j = 0; j < 8; j++) { int i = t + j * 256; p += xi[i] - xj[i]; }
    __shared__ float red[256];
    red[t] = p;
    __syncthreads();
    for (int s = 128; s > 0; s >>= 1) { if (t < s) red[t] += red[t + s]; __syncthreads(); }
    if (t == 0) pb[e] = red[0];
}

// ---------- K9: final deterministic reduction -> scalar ----------
__global__ __launch_bounds__(256) void final_kernel(const float* __restrict__ pb,
                                                    const float* __restrict__ lh,
                                                    float* __restrict__ out_scalar) {
    __shared__ float red[256];
    __shared__ float dmean;
    const int t = threadIdx.x;
    float s = 0.0f;
    for (int i = t; i < kE; i += 256) s += pb[i];
    red[t] = s;
    __syncthreads();
    for (int st = 128; st > 0; st >>= 1) { if (t < st) red[t] += red[t + st]; __syncthreads(); }
    if (t == 0) dmean = red[0] / (float)((size_t)kE * kB * kC);
    __syncthreads();
    float s2 = 0.0f;
    for (int i = t; i < kM * kM; i += 256) s2 += fabsf(lh[i]);
    red[t] = s2;
    __syncthreads();
    for (int st = 128; st > 0; st >>= 1) { if (t < st) red[t] += red[t + st]; __syncthreads(); }
    if (t == 0) out_scalar[0] = fabsf(dmean) + red[0] / (float)((kM + 1) * (kM + 1));
}

extern "C" void kernel_launch(void* const* d_in, const int* in_sizes, int n_in,
                              void* d_out, int out_size, void* d_ws, size_t ws_size,
                              hipStream_t stream) {
    const float* x   = (const float*)d_in[0];
    const int* nodes = (const int*)d_in[1];
    const int* edges = (const int*)d_in[2];
    const float* w   = (const float*)d_in[3];
    const float* att = (const float*)d_in[4];
    float* out = (float*)d_out;

    float* ws = (float*)d_ws;
    float* X1 = ws;                                   // [N,B,C] = 8388608 f
    float* ES = X1 + (size_t)kN * kBC;                // [M,B,C] = 131072 f
    float* OE = ES + (size_t)kM * kBC;                // [M,B,C] = 131072 f
    float* Y1 = OE + (size_t)kM * kBC;                // [N,B]
    float* Y2 = Y1 + kNB;                             // [M,B]
    float* SQ = Y2 + kMB;                             // [M,B]
    float* ALPHA = SQ + kMB;                          // [E,B]
    unsigned* SEGMAX = (unsigned*)(ALPHA + (size_t)kE * kB);  // [N,B]
    float* SEGSUM = (float*)(SEGMAX + kNB);           // [N,B]
    float* PB = SEGSUM + kNB;                         // [E]
    float* LH = PB + kE;                              // [M*M]
    int* DEG = (int*)(LH + kM * kM);                  // [N]

    init_kernel<<<(kB * kN * kC + 255) / 256, 256, 0, stream>>>(out, SEGMAX, SEGSUM, DEG);
    gemm_xw_wmma<<<(kB * kN) / (16 * 8), 256, 0, stream>>>(x, w, X1);          // 512 blocks
    edge_sums_kernel<<<kM, 256, 0, stream>>>(X1, nodes, edges, ES);
    dots_kernel<<<(kNB + 2 * kMB) / 8, 256, 0, stream>>>(X1, ES, att, Y1, Y2, SQ);
    alpha_logits_kernel<<<(kE * kB) / 256, 256, 0, stream>>>(nodes, edges, Y1, Y2, ALPHA, SEGMAX, DEG);
    alpha_exp_kernel<<<(kE * kB) / 256, 256, 0, stream>>>(nodes, ALPHA, SEGMAX, SEGSUM);
    alpha_norm_kernel<<<(kE * kB) / 256, 256, 0, stream>>>(nodes, ALPHA, SEGSUM);
    out_e_kernel<<<kM, 256, 0, stream>>>(X1, nodes, edges, ALPHA, OE);
    scatter_out_kernel<<<(kE * kB * kC) / 256, 256, 0, stream>>>(nodes, edges, ALPHA, OE, DEG, out);
    pair_loss_kernel<<<kM * kM, 128, 0, stream>>>(ES, SQ, LH);
    diff_kernel<<<kE, 256, 0, stream>>>(X1, ES, nodes, edges, PB);
    final_kernel<<<1, 256, 0, stream>>>(PB, LH, out + (size_t)kB * kN * kC);
}